// AttnInteractionLayer_2851858284689
// MI455X (gfx1250) — compile-verified
//
#include <hip/hip_runtime.h>
#include <hip/hip_bf16.h>
#include <cstdint>

// Problem constants (from the reference)
#define RR     4096
#define FF     32
#define KDIM   256          // IN
#define NDIM   512          // OUT*H
#define MTOT   (RR * FF)    // 131072 GEMM rows

// Tiling
#define M_WG   64           // rows per workgroup
#define SA     260          // LDS row stride (floats) for X tile: 16B-aligned, bank-staggered
#define SY     516          // LDS row stride (floats) for y tile
#define Y_ELEMS (M_WG * SY) // 33024 floats

typedef __attribute__((ext_vector_type(2))) float v2f;
typedef __attribute__((ext_vector_type(4))) float v4f;
typedef __attribute__((ext_vector_type(8))) float v8f;

// ---------------------------------------------------------------------------
// Kernel 1: fold W_c = W_v + W_r into a K-pair-interleaved layout:
//   Wcp[p][n] = float2( Wc[2p][n], Wc[2p+1][n] ),  p = 0..127, n = 0..511
// so a WMMA B fragment is one b64 load per lane instead of two b32 loads.
// Flat float index: p*1024 + n*2 + (k&1)
// ---------------------------------------------------------------------------
__global__ __launch_bounds__(256) void fold_weights_kernel(
    const float* __restrict__ Wv, const float* __restrict__ Wr,
    float* __restrict__ Wcp, int n)
{
    int i = blockIdx.x * blockDim.x + threadIdx.x;
    if (i < n) {
        int k = i >> 9;          // 0..255
        int c = i & 511;         // 0..511
        float v = Wv[i] + Wr[i];
        Wcp[(size_t)(k >> 1) * 1024 + c * 2 + (k & 1)] = v;
    }
}

// ---------------------------------------------------------------------------
// Kernel 2: fused GEMM (f32 WMMA 16x16x4) + leaky_relu + layernorm
//   X   : (MTOT, 256) f32
//   Wcp : K-pair-interleaved (128, 512, 2) f32 (L2-resident, 512 KB)
//   out : (MTOT, 512) f32
// Workgroup: 256 threads = 8 wave32. WG tile 64x512; wave tile 64x64.
// ---------------------------------------------------------------------------
__global__ __launch_bounds__(256) void attn_gemm_ln_kernel(
    const float* __restrict__ X,
    const float* __restrict__ Wcp,
    const float* __restrict__ gamma,
    const float* __restrict__ beta,
    float* __restrict__ out)
{
    extern __shared__ float smem[];
    float* Alds   = smem;                 // 64 x SA floats (aliases ylds; barrier-separated)
    float* ylds   = smem;                 // 64 x SY floats
    float* psum   = smem + Y_ELEMS;       // 64 x 4
    float* psumsq = psum + 256;           // 64 x 4
    float* meanb  = psumsq + 256;         // 64
    float* rstdb  = meanb + 64;           // 64

    const int tid   = threadIdx.x;
    const int lane  = tid & 31;
    const int wave  = tid >> 5;           // 0..7 -> N column group
    const int l16   = lane & 15;
    const int lh    = lane >> 4;          // 0 or 1 (selects K pair within WMMA frag)
    const int rowWG = blockIdx.x * M_WG;  // first GEMM row of this WG
    const int colW  = wave * 64;          // first N column of this wave

    // ---- cooperative load of X tile (64 x 256 f32) into LDS, float4 wide ----
    {
        const v4f* __restrict__ Xg = (const v4f*)(X + (size_t)rowWG * KDIM);
        #pragma unroll
        for (int i = 0; i < 16; ++i) {
            int idx = i * 256 + tid;           // 0 .. 4095 float4s
            int row = idx >> 6;                // /64 float4 per row
            int c4  = idx & 63;
            v4f v = Xg[(size_t)row * 64 + c4];
            *(v4f*)&Alds[row * SA + c4 * 4] = v;
        }
    }
    __syncthreads();

    // ---- K loop: 64 steps of K=4, 16 WMMA tiles per wave per step ----
    v8f acc[4][4];
    #pragma unroll
    for (int mt = 0; mt < 4; ++mt)
        #pragma unroll
        for (int nt = 0; nt < 4; ++nt)
            acc[mt][nt] = (v8f){0.f, 0.f, 0.f, 0.f, 0.f, 0.f, 0.f, 0.f};

    // Per-lane base into the interleaved weights: pair index (k0/2 + lh), column n
    const float* WcpLane = Wcp + (size_t)lh * 1024 + (size_t)(colW + l16) * 2;

    #pragma unroll 4
    for (int k0 = 0; k0 < KDIM; k0 += 4) {
        // A fragments: 16x4 f32 layout -> lane l: M = l&15, K pair = 2*(l>>4)
        v2f a[4];
        #pragma unroll
        for (int mt = 0; mt < 4; ++mt)
            a[mt] = *(const v2f*)&Alds[(mt * 16 + l16) * SA + k0 + 2 * lh];

        // B fragments: 4x16 f32 layout -> lane l: N = l&15, K pair = 2*(l>>4)
        // One b64 load per tile from the K-pair-interleaved weights.
        v2f b[4];
        #pragma unroll
        for (int nt = 0; nt < 4; ++nt)
            b[nt] = *(const v2f*)(WcpLane + (size_t)(k0 >> 1) * 1024 + nt * 32);

        #pragma unroll
        for (int mt = 0; mt < 4; ++mt)
            #pragma unroll
            for (int nt = 0; nt < 4; ++nt)
                acc[mt][nt] = __builtin_amdgcn_wmma_f32_16x16x4_f32(
                    false, a[mt], false, b[nt], (short)0, acc[mt][nt], false, false);
    }

    // ---- epilogue: leaky_relu into LDS (reuses A region) ----
    __syncthreads();   // all waves done reading Alds
    #pragma unroll
    for (int mt = 0; mt < 4; ++mt) {
        #pragma unroll
        for (int nt = 0; nt < 4; ++nt) {
            #pragma unroll
            for (int j = 0; j < 8; ++j) {
                int row = mt * 16 + j + 8 * lh;      // C/D layout: VGPR j -> M=j (lanes 0-15), M=j+8 (lanes 16-31)
                int col = colW + nt * 16 + l16;
                float v = acc[mt][nt][j];
                ylds[row * SY + col] = (v >= 0.f) ? v : 0.01f * v;
            }
        }
    }
    __syncthreads();

    // ---- per-row partial reduction: 4 threads per row, 128 cols each ----
    {
        int row = tid >> 2;
        int q   = tid & 3;
        const float* yr = &ylds[row * SY + q * 128];
        float s = 0.f, ss = 0.f;
        #pragma unroll 8
        for (int c = 0; c < 128; ++c) {
            float v = yr[c];
            s  += v;
            ss += v * v;
        }
        psum[row * 4 + q]   = s;
        psumsq[row * 4 + q] = ss;
    }
    __syncthreads();

    if (tid < 64) {
        float s  = psum[tid * 4] + psum[tid * 4 + 1] + psum[tid * 4 + 2] + psum[tid * 4 + 3];
        float ss = psumsq[tid * 4] + psumsq[tid * 4 + 1] + psumsq[tid * 4 + 2] + psumsq[tid * 4 + 3];
        float mean = s * (1.0f / NDIM);
        float var  = ss * (1.0f / NDIM) - mean * mean;
        meanb[tid] = mean;
        rstdb[tid] = __frsqrt_rn(var + 1e-5f);
    }
    __syncthreads();

    // ---- normalize + affine + coalesced store ----
    {
        float* outg = out + (size_t)rowWG * NDIM;
        #pragma unroll 4
        for (int i = 0; i < 128; ++i) {
            int idx = i * 256 + tid;              // 0 .. 32767
            int row = idx >> 9;
            int col = idx & 511;
            float v = ylds[row * SY + col];
            float r = (v - meanb[row]) * rstdb[row] * gamma[col] + beta[col];
            outg[(size_t)row * NDIM + col] = r;
        }
    }
}

// ---------------------------------------------------------------------------
// Harness entry point
// Inputs (setup_inputs order): 0:x 1:W_q 2:W_k 3:W_v 4:W_r 5:ln_gamma 6:ln_beta
// ---------------------------------------------------------------------------
extern "C" void kernel_launch(void* const* d_in, const int* in_sizes, int n_in,
                              void* d_out, int out_size, void* d_ws, size_t ws_size,
                              hipStream_t stream)
{
    const float* x     = (const float*)d_in[0];
    const float* Wv    = (const float*)d_in[3];
    const float* Wr    = (const float*)d_in[4];
    const float* gamma = (const float*)d_in[5];
    const float* beta  = (const float*)d_in[6];
    float* out = (float*)d_out;
    float* Wcp = (float*)d_ws;            // 256*512 f32 = 512 KB scratch (interleaved)

    const int wn = KDIM * NDIM;           // 131072
    fold_weights_kernel<<<(wn + 255) / 256, 256, 0, stream>>>(Wv, Wr, Wcp, wn);

    const size_t lds_bytes = (size_t)(Y_ELEMS + 256 + 256 + 64 + 64) * sizeof(float); // ~134.5 KB
    attn_gemm_ln_kernel<<<MTOT / M_WG, 256, lds_bytes, stream>>>(x, Wcp, gamma, beta, out);
}